// VanillaBellmanFordLayer_3745211482942
// MI455X (gfx1250) — compile-verified
//
#include <hip/hip_runtime.h>

// Problem constants (from reference)
#define B_ 2
#define N_ 50000
#define D_ 128
#define E_ 800000
#define BN_ (B_ * N_)
#define LN_EPS 1e-5f

// GEMM tiling
#define ROWS 64        // rows (M) per block = 4 m-tiles of 16
#define PADK 132       // 128 + 4 -> conflict-free LDS (bank = 4*row + k)
#define K2 256
#define PADD 132

typedef __attribute__((ext_vector_type(2))) float v2f;
typedef __attribute__((ext_vector_type(8))) float v8f;

// Low 32 bits of a flat shared-aperture address == LDS byte offset (ISA §10.2:
// aperture base lives in addr[63:32], LDS offset is the low 32 bits).
__device__ __forceinline__ unsigned lds_off(const void* p) {
  return (unsigned)(unsigned long long)(size_t)p;
}

// Async DMA: 16 bytes global -> LDS, tracked by ASYNCcnt (no VGPR round-trip).
__device__ __forceinline__ void async_b128(unsigned lds_byte, const void* gsrc) {
  asm volatile("global_load_async_to_lds_b128 %0, %1, off"
               :: "v"(lds_byte), "v"(gsrc) : "memory");
}
__device__ __forceinline__ void wait_async0() {
  asm volatile("s_wait_asynccnt 0" ::: "memory");
}

// Native fp32 RMW at L2 atomic units, device scope, no return (STOREcnt).
__device__ __forceinline__ void atomic_add_f32(float* p, float v) {
  asm volatile("global_atomic_add_f32 %0, %1, off scope:SCOPE_DEV"
               :: "v"(p), "v"(v) : "memory");
}

// ---------------------------------------------------------------- zero agg
__global__ void zero_kernel(float4* __restrict__ p, int n4) {
  int i = blockIdx.x * blockDim.x + threadIdx.x;
  int stride = gridDim.x * blockDim.x;
  float4 z; z.x = z.y = z.z = z.w = 0.f;
  for (; i < n4; i += stride) p[i] = z;
}

// ------------------------------------------------- hm = h @ W_msg^T  (K=128)
// hm[m, f] = sum_d h[m, d] * Wm[f, d],  m in [0, B*N)
__global__ __launch_bounds__(256)
void gemm_msg_kernel(const float* __restrict__ h, const float* __restrict__ Wm,
                     float* __restrict__ hm) {
  __shared__ float sW[D_ * PADK];    // W_msg [f][d], ~66 KB
  __shared__ float sA[ROWS * PADK];  // h rows [r][d], ~33 KB
  const int tid  = threadIdx.x;
  const int wave = tid >> 5;
  const int lane = tid & 31;
  const int m0   = blockIdx.x * ROWS;

  // Stage via async global->LDS DMA (float4 granules; pad never splits a granule)
  const unsigned sW0 = lds_off(sW);
  for (int i = tid; i < (D_ * D_) / 4; i += 256) {       // 16 iters, no divergence
    int r = i >> 5, c = (i & 31) * 4;
    async_b128(sW0 + (unsigned)((r * PADK + c) * 4), Wm + i * 4);
  }
  const unsigned sA0 = lds_off(sA);
  for (int i = tid; i < (ROWS * D_) / 4; i += 256) {     // 8 iters
    int r = i >> 5, c = (i & 31) * 4;
    int m = m0 + r; if (m >= BN_) m = BN_ - 1;           // clamp tail (stores guarded)
    async_b128(sA0 + (unsigned)((r * PADK + c) * 4), h + (size_t)m * D_ + c);
  }
  wait_async0();
  __syncthreads();

  const int half = lane >> 4;       // 0: K+{0,1}, 1: K+{2,3}
  const int l16  = lane & 15;
  const int fcol = wave * 16 + l16; // this wave's output column

  v8f acc[4] = {};
  for (int k0 = 0; k0 < D_; k0 += 4) {
    const int kb = k0 + half * 2;
    v2f b;
    b.x = sW[fcol * PADK + kb];
    b.y = sW[fcol * PADK + kb + 1];
#pragma unroll
    for (int t = 0; t < 4; ++t) {
      v2f a;
      a.x = sA[(t * 16 + l16) * PADK + kb];
      a.y = sA[(t * 16 + l16) * PADK + kb + 1];
      acc[t] = __builtin_amdgcn_wmma_f32_16x16x4_f32(
          false, a, false, b, (short)0, acc[t], false, false);
    }
  }

#pragma unroll
  for (int t = 0; t < 4; ++t) {
    const float* cf = (const float*)&acc[t];
#pragma unroll
    for (int v = 0; v < 8; ++v) {
      int ml = t * 16 + v + half * 8;   // C/D layout: M = v + 8*half
      int gm = m0 + ml;
      if (gm < BN_) hm[(size_t)gm * D_ + fcol] = cf[v];
    }
  }
}

// -------------------------------------- edge gather * rel, scatter-add (atomics)
__global__ __launch_bounds__(256)
void edge_kernel(const float* __restrict__ hm, const float* __restrict__ rel,
                 const int* __restrict__ esrc, const int* __restrict__ etgt,
                 const int* __restrict__ erel, float* __restrict__ agg) {
  long long tid = (long long)blockIdx.x * blockDim.x + threadIdx.x;
  int e = (int)(tid >> 5);
  if (e >= E_) return;
  int q = (int)(tid & 31);                 // 4 floats per lane
  // One edge per wave: indices are wave-uniform -> force scalar regs.
  int src = __builtin_amdgcn_readfirstlane(esrc[e]);
  int tgt = __builtin_amdgcn_readfirstlane(etgt[e]);
  int rl  = __builtin_amdgcn_readfirstlane(erel[e]);
  const float4 rv = ((const float4*)rel)[rl * 32 + q];
#pragma unroll
  for (int b = 0; b < B_; ++b) {
    float4 hv = ((const float4*)hm)[(size_t)(b * N_ + src) * 32 + q];
    float* dst = agg + (size_t)(b * N_ + tgt) * D_ + q * 4;
    atomic_add_f32(dst + 0, hv.x * rv.x);
    atomic_add_f32(dst + 1, hv.y * rv.y);
    atomic_add_f32(dst + 2, hv.z * rv.z);
    atomic_add_f32(dst + 3, hv.w * rv.w);
  }
}

// ------------- upd = relu([h|agg] @ W_upd^T + b); out = LN(h + upd)  (K=256)
__global__ __launch_bounds__(256)
void update_kernel(const float* __restrict__ h, const float* __restrict__ agg,
                   const float* __restrict__ Wu, const float* __restrict__ bu,
                   const float* __restrict__ lscale, const float* __restrict__ lbias,
                   float* __restrict__ out) {
  __shared__ float sAh[ROWS * PADK];  // h   part of cat, ~33 KB
  __shared__ float sAg[ROWS * PADK];  // agg part of cat, ~33 KB
  __shared__ float sU[ROWS * PADD];   // raw upd [r][f],  ~33 KB
  const int tid  = threadIdx.x;
  const int wave = tid >> 5;
  const int lane = tid & 31;
  const int m0   = blockIdx.x * ROWS;

  const unsigned sAh0 = lds_off(sAh);
  const unsigned sAg0 = lds_off(sAg);
  for (int i = tid; i < (ROWS * D_) / 4; i += 256) {     // 8 iters each
    int r = i >> 5, c = (i & 31) * 4;
    int m = m0 + r; if (m >= BN_) m = BN_ - 1;
    unsigned doff = (unsigned)((r * PADK + c) * 4);
    async_b128(sAh0 + doff, h   + (size_t)m * D_ + c);
    async_b128(sAg0 + doff, agg + (size_t)m * D_ + c);
  }
  wait_async0();
  __syncthreads();

  const int half = lane >> 4;
  const int l16  = lane & 15;
  const int fcol = wave * 16 + l16;

  v8f acc[4] = {};
  for (int k0 = 0; k0 < K2; k0 += 4) {
    const int kbW = k0 + half * 2;           // 0..255, into W_upd rows
    const int kb  = (k0 & 127) + half * 2;   // 0..127, into sAh/sAg
    const float* sAk = (k0 < D_) ? sAh : sAg;
    v2f b;                                   // W_upd panel: 128 KB, L2/L1 resident
    b.x = Wu[fcol * K2 + kbW];
    b.y = Wu[fcol * K2 + kbW + 1];
#pragma unroll
    for (int t = 0; t < 4; ++t) {
      v2f a;
      a.x = sAk[(t * 16 + l16) * PADK + kb];
      a.y = sAk[(t * 16 + l16) * PADK + kb + 1];
      acc[t] = __builtin_amdgcn_wmma_f32_16x16x4_f32(
          false, a, false, b, (short)0, acc[t], false, false);
    }
  }

#pragma unroll
  for (int t = 0; t < 4; ++t) {
    const float* cf = (const float*)&acc[t];
#pragma unroll
    for (int v = 0; v < 8; ++v) {
      int ml = t * 16 + v + half * 8;
      sU[ml * PADD + fcol] = cf[v];
    }
  }
  __syncthreads();

  // LayerNorm: each wave handles 8 rows; lane covers 4 columns
  for (int rr = 0; rr < ROWS; rr += 8) {
    int r = rr + wave;
    int gm = m0 + r;
    if (gm >= BN_) continue;            // uniform across the wave
    float x[4], s = 0.f, ss = 0.f;
#pragma unroll
    for (int j = 0; j < 4; ++j) {
      int c = lane * 4 + j;
      float u = sU[r * PADD + c] + bu[c];
      u = u > 0.f ? u : 0.f;            // relu
      float xv = h[(size_t)gm * D_ + c] + u;  // residual
      x[j] = xv; s += xv; ss += xv * xv;
    }
#pragma unroll
    for (int off = 16; off >= 1; off >>= 1) {
      s  += __shfl_xor(s,  off, 32);
      ss += __shfl_xor(ss, off, 32);
    }
    float mean = s * (1.f / D_);
    float var  = ss * (1.f / D_) - mean * mean;
    float rstd = rsqrtf(var + LN_EPS);
#pragma unroll
    for (int j = 0; j < 4; ++j) {
      int c = lane * 4 + j;
      out[(size_t)gm * D_ + c] = (x[j] - mean) * rstd * lscale[c] + lbias[c];
    }
  }
}

// ---------------------------------------------------------------- launcher
extern "C" void kernel_launch(void* const* d_in, const int* in_sizes, int n_in,
                              void* d_out, int out_size, void* d_ws, size_t ws_size,
                              hipStream_t stream) {
  const float* h   = (const float*)d_in[0];
  const float* Wm  = (const float*)d_in[1];
  const float* rel = (const float*)d_in[2];
  const float* Wu  = (const float*)d_in[3];
  const float* bu  = (const float*)d_in[4];
  const float* ls  = (const float*)d_in[5];
  const float* lb  = (const float*)d_in[6];
  const int* esrc  = (const int*)d_in[7];
  const int* etgt  = (const int*)d_in[8];
  const int* erel  = (const int*)d_in[9];
  (void)in_sizes; (void)n_in; (void)out_size; (void)ws_size;

  float* outp = (float*)d_out;
  float* hm   = (float*)d_ws;                 // [B*N, D] = 51.2 MB
  float* agg  = hm + (size_t)BN_ * D_;        // [B*N, D] = 51.2 MB

  const int n4 = (BN_ * D_) / 4;
  zero_kernel<<<2048, 256, 0, stream>>>((float4*)agg, n4);

  const int gemm_blocks = (BN_ + ROWS - 1) / ROWS;  // 1563
  gemm_msg_kernel<<<gemm_blocks, 256, 0, stream>>>(h, Wm, hm);

  const long long ethreads = (long long)E_ * 32;
  edge_kernel<<<(int)((ethreads + 255) / 256), 256, 0, stream>>>(
      hm, rel, esrc, etgt, erel, agg);

  update_kernel<<<gemm_blocks, 256, 0, stream>>>(h, agg, Wu, bu, ls, lb, outp);
}